// CCMambaEncoder_19129784336936
// MI455X (gfx1250) — compile-verified
//
#include <hip/hip_runtime.h>
#include <hip/hip_bf16.h>
#include <math.h>

// ---------------------------------------------------------------------------
// CCMambaEncoder forward for MI455X (gfx1250, wave32, WMMA).
// Dominant cost: 3x (8192x8192)@(8192x128) fp32 GEMMs -> HBM-bound (~768MB of
// adjacency streaming @ 23.3 TB/s). GEMM uses V_WMMA_F32_16X16X4_F32 with
// double-buffered LDS; the A (adjacency) tile is staged with the CDNA5 async
// DMA path (global_load_async_to_lds_b128 + s_wait_asynccnt).
// ---------------------------------------------------------------------------

#define NN     8192
#define DD     128
#define DIN    256      // D_INNER
#define DSTATE 16
#define DTRANK 8

typedef float v2f __attribute__((ext_vector_type(2)));
typedef float v8f __attribute__((ext_vector_type(8)));

#define ACT_NONE     0
#define ACT_RELU     1
#define ACT_SOFTPLUS 2

__device__ __forceinline__ float sigmoidf_(float x) { return 1.f / (1.f + __expf(-x)); }

// ---------------------------------------------------------------------------
// WMMA GEMM: C[M x Nout] = act(A[M x K] * B + bias)
//   b_is_wt==1 : B is W (Nout x ldb==K) row-major; logical B[k][n] = W[n*ldb+k]
//   b_is_wt==0 : B row-major (K x ldb);            logical B[k][n] = B[k*ldb+n]
// Tile: 128(M) x 128(N) per block (8 waves), K in 32-chunks, double-buffered.
//   A tile: LDS row-major [m][36] (padded; 16B rows -> async B128 staging,
//           and ds_load_b64 A-fragment reads are bank-conflict-free).
//   B tile: LDS paired-k [k/2][n][2] so a B fragment is one ds_load_b64.
// Fragments per CDNA5 ISA 7.12.2:
//   A 16x4 f32: lanes 0-15 K={0,1}, lanes 16-31 K={2,3} of each k4 group
//   C  16x16  : VGPR j -> rows j (lanes 0-15) / j+8 (lanes 16-31)
// ---------------------------------------------------------------------------
__global__ __launch_bounds__(256)
void gemm_wmma_f32(const float* __restrict__ A, int lda,
                   const float* __restrict__ B, int ldb, int b_is_wt,
                   const float* __restrict__ bias,
                   float* __restrict__ C, int ldc,
                   int M, int K, int Nout, int act)
{
    __shared__ __attribute__((aligned(16))) float As[2][128][36];
    __shared__ __attribute__((aligned(16))) float Bs[2][16][128][2];

    const int tid  = threadIdx.x;
    const int wave = tid >> 5;
    const int lane = tid & 31;
    const int l15  = lane & 31 & 15;
    const int koff = (lane >> 4) << 1;          // 0 or 2

    v8f acc[8];
    #pragma unroll
    for (int i = 0; i < 8; ++i) acc[i] = (v8f){0.f,0.f,0.f,0.f,0.f,0.f,0.f,0.f};

    const int mA       = tid & 127;
    const int qA0      = (tid >> 7) * 4;        // 0 or 4
    const int grow_blk = blockIdx.x * 128;
    const int gcol_blk = blockIdx.y * 128;
    const bool mfull   = (grow_blk + 128) <= M;

    // ---- A tile staging: async DMA when the chunk is fully in-bounds ----
    auto stageA = [&](int buf, int kb) -> bool {
        bool full = mfull && (kb + 32 <= K);
        int  grow = grow_blk + mA;
        if (full) {
            #pragma unroll
            for (int qq = 0; qq < 4; ++qq) {
                int q = qA0 + qq;
                unsigned lds = (unsigned)(size_t)&As[buf][mA][q * 4];
                unsigned long long ga =
                    (unsigned long long)(size_t)(A + (size_t)grow * lda + kb + q * 4);
                asm volatile("global_load_async_to_lds_b128 %0, %1, off"
                             :: "v"(lds), "v"(ga) : "memory");
            }
        } else {
            #pragma unroll
            for (int qq = 0; qq < 4; ++qq) {
                int q  = qA0 + qq;
                int kk = kb + q * 4;
                float4 v = make_float4(0.f, 0.f, 0.f, 0.f);
                if (grow < M) {
                    const float* ap = A + (size_t)grow * lda;
                    if (kk + 3 < K) v = *(const float4*)(ap + kk);
                    else {
                        if (kk + 0 < K) v.x = ap[kk + 0];
                        if (kk + 1 < K) v.y = ap[kk + 1];
                        if (kk + 2 < K) v.z = ap[kk + 2];
                        if (kk + 3 < K) v.w = ap[kk + 3];
                    }
                }
                *(float4*)&As[buf][mA][q * 4] = v;
            }
        }
        return full;
    };

    // ---- B tile: register fetch + paired-k LDS store ----
    auto fetchB = [&](int kb, float4* r) {
        if (b_is_wt) {
            int nB = tid & 127, q0 = (tid >> 7) * 4;
            int gcol = gcol_blk + nB;
            #pragma unroll
            for (int qq = 0; qq < 4; ++qq) {
                int kk = kb + (q0 + qq) * 4;
                float4 v = make_float4(0.f, 0.f, 0.f, 0.f);
                if (gcol < Nout) {
                    const float* bp = B + (size_t)gcol * ldb;
                    if (kk + 3 < K) v = *(const float4*)(bp + kk);
                    else {
                        if (kk + 0 < K) v.x = bp[kk + 0];
                        if (kk + 1 < K) v.y = bp[kk + 1];
                        if (kk + 2 < K) v.z = bp[kk + 2];
                        if (kk + 3 < K) v.w = bp[kk + 3];
                    }
                }
                r[qq] = v;
            }
        } else {
            int  kB  = tid & 31;
            int  cb  = (tid >> 5) * 16;
            bool kok = (kb + kB) < K;
            const float* bp = B + (size_t)(kb + kB) * ldb;
            #pragma unroll
            for (int qq = 0; qq < 4; ++qq) {
                int gcol = gcol_blk + cb + qq * 4;
                float4 v = make_float4(0.f, 0.f, 0.f, 0.f);
                if (kok) {
                    if (gcol + 3 < Nout) v = *(const float4*)(bp + gcol);
                    else {
                        if (gcol + 0 < Nout) v.x = bp[gcol + 0];
                        if (gcol + 1 < Nout) v.y = bp[gcol + 1];
                        if (gcol + 2 < Nout) v.z = bp[gcol + 2];
                        if (gcol + 3 < Nout) v.w = bp[gcol + 3];
                    }
                }
                r[qq] = v;
            }
        }
    };
    auto storeB = [&](int buf, const float4* r) {
        if (b_is_wt) {
            int nB = tid & 127, q0 = (tid >> 7) * 4;
            #pragma unroll
            for (int qq = 0; qq < 4; ++qq) {
                int q = q0 + qq;
                v2f lo; lo.x = r[qq].x; lo.y = r[qq].y;
                v2f hi; hi.x = r[qq].z; hi.y = r[qq].w;
                *(v2f*)&Bs[buf][2 * q    ][nB][0] = lo;
                *(v2f*)&Bs[buf][2 * q + 1][nB][0] = hi;
            }
        } else {
            int kB = tid & 31, cb = (tid >> 5) * 16;
            int k2 = kB >> 1, par = kB & 1;
            #pragma unroll
            for (int qq = 0; qq < 4; ++qq) {
                int nc = cb + qq * 4;
                Bs[buf][k2][nc + 0][par] = r[qq].x;
                Bs[buf][k2][nc + 1][par] = r[qq].y;
                Bs[buf][k2][nc + 2][par] = r[qq].z;
                Bs[buf][k2][nc + 3][par] = r[qq].w;
            }
        }
    };

    // ---- prologue: stage chunk 0 ----
    float4 rb[4];
    bool pend = stageA(0, 0);
    fetchB(0, rb);
    storeB(0, rb);
    if (pend) asm volatile("s_wait_asynccnt 0x0" ::: "memory");
    __syncthreads();

    const int nchunks = (K + 31) >> 5;
    const int mrow    = wave * 16 + l15;
    for (int c = 0; c < nchunks; ++c) {
        const int  cur  = c & 1;
        const int  nxt  = cur ^ 1;
        const bool more = (c + 1) < nchunks;
        bool pa = false;
        if (more) {                       // start next-chunk A DMA + B fetch early
            pa = stageA(nxt, (c + 1) << 5);
            fetchB((c + 1) << 5, rb);
        }
        // ---------------- WMMA compute on current buffers ----------------
        #pragma unroll
        for (int kk8 = 0; kk8 < 8; ++kk8) {
            int k4 = kk8 * 4 + koff;
            int k2 = kk8 * 2 + (lane >> 4);
            v2f a = *(const v2f*)&As[cur][mrow][k4];
            #pragma unroll
            for (int nt = 0; nt < 8; ++nt) {
                v2f b = *(const v2f*)&Bs[cur][k2][nt * 16 + l15][0];
                acc[nt] = __builtin_amdgcn_wmma_f32_16x16x4_f32(
                    false, a, false, b, (short)0, acc[nt], false, false);
            }
        }
        if (more) storeB(nxt, rb);
        if (pa) asm volatile("s_wait_asynccnt 0x0" ::: "memory");
        __syncthreads();
    }

    // ---------------- epilogue: bias + act + store ----------------
    #pragma unroll
    for (int nt = 0; nt < 8; ++nt) {
        int gcol = gcol_blk + nt * 16 + l15;
        if (gcol >= Nout) continue;
        float bv = bias ? bias[gcol] : 0.f;
        #pragma unroll
        for (int j = 0; j < 8; ++j) {
            int grow = grow_blk + wave * 16 + j + ((lane >> 4) << 3);
            if (grow >= M) continue;
            float v = acc[nt][j] + bv;
            if (act == ACT_RELU)          v = fmaxf(v, 0.f);
            else if (act == ACT_SOFTPLUS) v = (v > 20.f) ? v : log1pf(__expf(v));
            C[(size_t)grow * ldc + gcol] = v;
        }
    }
}

// --------------------------- small helper kernels ---------------------------

// out[m][n] = act(sum_k A[m][k]*W[n*K+k] + b[n]) ; used only for Nout==1 cases
__global__ void linear_naive(const float* __restrict__ A, int lda,
                             const float* __restrict__ W, const float* __restrict__ bias,
                             float* __restrict__ C, int ldc, int M, int K, int Nout, int act)
{
    int t = blockIdx.x * blockDim.x + threadIdx.x;
    if (t >= M * Nout) return;
    int m = t / Nout, n = t % Nout;
    float s = bias ? bias[n] : 0.f;
    const float* ap = A + (size_t)m * lda;
    const float* wp = W + (size_t)n * K;
    for (int k = 0; k < K; ++k) s += ap[k] * wp[k];
    if (act == ACT_RELU) s = fmaxf(s, 0.f);
    C[(size_t)m * ldc + n] = s;
}

// causal depthwise conv (taps=4) + SiLU; reads xc-half of xz (cols 0..255 of 512)
__global__ void conv_silu_k(const float* __restrict__ xz, const float* __restrict__ cw,
                            const float* __restrict__ cb, float* __restrict__ xc, int L)
{
    int t = blockIdx.x * blockDim.x + threadIdx.x;
    if (t >= L * DIN) return;
    int row = t >> 8, d = t & 255;
    float s = cb[d];
    #pragma unroll
    for (int j = 0; j < 4; ++j) {
        int tt = row - 3 + j;
        if (tt >= 0) s += cw[d * 4 + j] * xz[(size_t)tt * 512 + d];
    }
    xc[(size_t)row * DIN + d] = s * sigmoidf_(s);
}

// Selective scan: one (d,s) channel per lane; 16-lane shuffle reduce per step.
__global__ __launch_bounds__(128)
void mamba_scan(const float* __restrict__ dt,   // L x 256
                const float* __restrict__ xdbl, // L x 40 (B @ col 8, C @ col 24)
                const float* __restrict__ xc,   // L x 256
                const float* __restrict__ xz,   // L x 512 (z @ col 256+d)
                const float* __restrict__ Alog, // 256 x 16
                const float* __restrict__ Dvec, // 256
                float* __restrict__ y,          // L x 256
                int L)
{
    int wave = blockIdx.x * (blockDim.x >> 5) + (threadIdx.x >> 5);
    int lane = threadIdx.x & 31;
    int s    = lane & 15;
    int d    = wave * 2 + (lane >> 4);
    float a  = -__expf(Alog[d * 16 + s]);
    float Dd = Dvec[d];
    float h  = 0.f;
    for (int t = 0; t < L; ++t) {
        float dtv = dt[(size_t)t * 256 + d];
        float xcv = xc[(size_t)t * 256 + d];
        float Bv  = xdbl[(size_t)t * 40 + 8 + s];
        float Cv  = xdbl[(size_t)t * 40 + 24 + s];
        h = h * __expf(dtv * a) + dtv * Bv * xcv;
        float p = h * Cv;
        p += __shfl_xor(p, 1, 32);
        p += __shfl_xor(p, 2, 32);
        p += __shfl_xor(p, 4, 32);
        p += __shfl_xor(p, 8, 32);
        if (s == 0) {
            float zv = xz[(size_t)t * 512 + 256 + d];
            float yv = p + Dd * xcv;
            yv *= zv * sigmoidf_(zv);
            y[(size_t)t * 256 + d] = yv;
        }
    }
}

// LayerNorm over 128 cols, one wave per row
__global__ void layernorm128(const float* __restrict__ in, float* __restrict__ out,
                             const float* __restrict__ w, const float* __restrict__ b, int rows)
{
    int wave = blockIdx.x * (blockDim.x >> 5) + (threadIdx.x >> 5);
    int lane = threadIdx.x & 31;
    if (wave >= rows) return;
    const float* x = in + (size_t)wave * 128;
    float4 v = *(const float4*)(x + lane * 4);
    float s = v.x + v.y + v.z + v.w;
    for (int m = 1; m < 32; m <<= 1) s += __shfl_xor(s, m, 32);
    float mu = s * (1.f / 128.f);
    float d0 = v.x - mu, d1 = v.y - mu, d2 = v.z - mu, d3 = v.w - mu;
    float q = d0 * d0 + d1 * d1 + d2 * d2 + d3 * d3;
    for (int m = 1; m < 32; m <<= 1) q += __shfl_xor(q, m, 32);
    float rinv = rsqrtf(q * (1.f / 128.f) + 1e-5f);
    float4 wv = *(const float4*)(w + lane * 4);
    float4 bv = *(const float4*)(b + lane * 4);
    float4 o;
    o.x = d0 * rinv * wv.x + bv.x;
    o.y = d1 * rinv * wv.y + bv.y;
    o.z = d2 * rinv * wv.z + bv.z;
    o.w = d3 * rinv * wv.w + bv.w;
    *(float4*)(out + (size_t)wave * 128 + lane * 4) = o;
}

__global__ void add_inplace(float* __restrict__ h, const float* __restrict__ a, int n)
{
    int i = blockIdx.x * blockDim.x + threadIdx.x;
    if (i < n) h[i] += a[i];
}

__global__ void init_idx(int* __restrict__ idx, int n)
{
    int i = blockIdx.x * blockDim.x + threadIdx.x;
    if (i < n) idx[i] = i;
}

__global__ void bitonic_step(float* __restrict__ keys, int* __restrict__ idx,
                             int n, int j, int k)
{
    int i = blockIdx.x * blockDim.x + threadIdx.x;
    if (i >= n) return;
    int l = i ^ j;
    if (l > i) {
        bool asc = ((i & k) == 0);
        float ki = keys[i], kl = keys[l];
        if ((ki > kl) == asc) {
            keys[i] = kl; keys[l] = ki;
            int ti = idx[i]; idx[i] = idx[l]; idx[l] = ti;
        }
    }
}

__global__ void gather_rows(float* __restrict__ dst, const float* __restrict__ src,
                            const int* __restrict__ idx, int n)
{
    int i = blockIdx.x * blockDim.x + threadIdx.x;
    if (i >= n * 128) return;
    int r = i >> 7, c = i & 127;
    dst[i] = src[(size_t)idx[r] * 128 + c];
}

__global__ void scatter_add_rows(float* __restrict__ h, const float* __restrict__ src,
                                 const int* __restrict__ idx, int n)
{
    int i = blockIdx.x * blockDim.x + threadIdx.x;
    if (i >= n * 128) return;
    int r = i >> 7, c = i & 127;
    h[(size_t)idx[r] * 128 + c] += src[i];
}

__global__ void concat2(float* __restrict__ cat, const float* __restrict__ a,
                        const float* __restrict__ b, int n)
{
    int i = blockIdx.x * blockDim.x + threadIdx.x;
    if (i >= n * 256) return;
    int r = i >> 8, c = i & 255;
    cat[i] = (c < 128) ? a[(size_t)r * 128 + c] : b[(size_t)r * 128 + (c - 128)];
}

__global__ void gate_fuse(float* __restrict__ out, const float* __restrict__ glin,
                          const float* __restrict__ hn, const float* __restrict__ hh, int n)
{
    int i = blockIdx.x * blockDim.x + threadIdx.x;
    if (i >= n) return;
    float g = sigmoidf_(glin[i]);
    out[i] = g * hn[i] + (1.f - g) * hh[i];
}

// ----------------------------- host-side driver -----------------------------

#define FP(i) ((const float*)d_in[(i)])

static inline void gemmF(hipStream_t st, const float* A, int lda, const float* B, int ldb,
                         int bwt, const float* bias, float* C, int ldc,
                         int M, int K, int Nout, int act)
{
    dim3 g((M + 127) / 128, (Nout + 127) / 128);
    gemm_wmma_f32<<<g, 256, 0, st>>>(A, lda, B, ldb, bwt, bias, C, ldc, M, K, Nout, act);
}

// mamba block params at d_in[base + 0..10]:
// in_proj.w, in_proj.b, conv_w, conv_b, x_proj_w, dt_proj.w, dt_proj.b, A_log, D,
// out_proj.w, out_proj.b
static void run_mamba(hipStream_t st, void* const* d_in, int base,
                      const float* in, float* out,
                      float* xz, float* xc, float* xdbl, float* dtb, float* yb)
{
    const int EB = 256, L = NN;
    gemmF(st, in, 128, FP(base + 0), 128, 1, FP(base + 1), xz, 512, L, 128, 512, ACT_NONE);
    conv_silu_k<<<(L * DIN + EB - 1) / EB, EB, 0, st>>>(xz, FP(base + 2), FP(base + 3), xc, L);
    gemmF(st, xc, 256, FP(base + 4), 256, 1, nullptr, xdbl, 40, L, 256, 40, ACT_NONE);
    gemmF(st, xdbl, 40, FP(base + 5), 8, 1, FP(base + 6), dtb, 256, L, 8, 256, ACT_SOFTPLUS);
    mamba_scan<<<32, 128, 0, st>>>(dtb, xdbl, xc, xz, FP(base + 7), FP(base + 8), yb, L);
    gemmF(st, yb, 256, FP(base + 9), 256, 1, FP(base + 10), out, 128, L, 256, 128, ACT_NONE);
}

extern "C" void kernel_launch(void* const* d_in, const int* in_sizes, int n_in,
                              void* d_out, int out_size, void* d_ws, size_t ws_size,
                              hipStream_t stream)
{
    (void)in_sizes; (void)n_in; (void)out_size; (void)ws_size;
    const int EB = 256;
    // input layout (setup_inputs insertion order, nested dicts in insertion order):
    // 0:x 1:adj_norm 2:adj_ho_norm 3:in0.w 4:in0.b 5:in1.w 6:in1.b
    // 7..21:local0  22..36:local1  37..51:global0  52..66:global1
    //   (block: mamba[11], norm_w, norm_b, extra.w, extra.b)
    // 67..74:ho(lin_node.w/b, lin_ho.w/b, gate.w/b, norm_w, norm_b)
    // 75..80: head0.w/b head1.w/b head2.w/b
    const int L0 = 7, L1 = 22, G0 = 37, G1 = 52, HO = 67;

    const float* X     = FP(0);
    const float* ADJ   = FP(1);
    const float* ADJHO = FP(2);

    float* ws = (float*)d_ws;
    size_t off = 0;
    auto alloc = [&](size_t n) { float* p = ws + off; off += n; return p; };
    float* h    = alloc((size_t)NN * 128);
    float* t0   = alloc((size_t)NN * 128);
    float* t1   = alloc((size_t)NN * 128);
    float* xz   = alloc((size_t)NN * 512);
    float* xc   = alloc((size_t)NN * 256);
    float* xdbl = alloc((size_t)NN * 40);
    float* dtb  = alloc((size_t)NN * 256);
    float* yb   = alloc((size_t)NN * 256);
    float* keys = alloc((size_t)NN);
    int*   idx  = (int*)alloc((size_t)NN);
    float* hn   = alloc((size_t)NN * 128);
    float* hh   = alloc((size_t)NN * 128);
    float* cat  = alloc((size_t)NN * 256);

    // ---- input embedding: h = lin1(relu(lin0(x))) ----
    gemmF(stream, X, 32, FP(3), 32, 1, FP(4), t0, 128, NN, 32, 128, ACT_RELU);
    gemmF(stream, t0, 128, FP(5), 128, 1, FP(6), h, 128, NN, 128, 128, ACT_NONE);

    // ---- local blocks: h += mamba(LN(relu(gcn(adj @ h)))) ----
    for (int pass = 0; pass < 2; ++pass) {
        int base = pass ? L1 : L0;
        gemmF(stream, ADJ, NN, h, 128, 0, nullptr, t0, 128, NN, NN, 128, ACT_NONE);
        gemmF(stream, t0, 128, FP(base + 13), 128, 1, FP(base + 14), t1, 128, NN, 128, 128, ACT_RELU);
        layernorm128<<<NN / 8, 256, 0, stream>>>(t1, t1, FP(base + 11), FP(base + 12), NN);
        run_mamba(stream, d_in, base, t1, t0, xz, xc, xdbl, dtb, yb);
        add_inplace<<<(NN * 128 + EB - 1) / EB, EB, 0, stream>>>(h, t0, NN * 128);
    }

    // ---- global blocks: sort by score, LN, mamba, scatter back ----
    for (int pass = 0; pass < 2; ++pass) {
        int base = pass ? G1 : G0;
        linear_naive<<<(NN + EB - 1) / EB, EB, 0, stream>>>(
            h, 128, FP(base + 13), FP(base + 14), keys, 1, NN, 128, 1, ACT_NONE);
        init_idx<<<(NN + EB - 1) / EB, EB, 0, stream>>>(idx, NN);
        for (int k = 2; k <= NN; k <<= 1)
            for (int j = k >> 1; j > 0; j >>= 1)
                bitonic_step<<<(NN + EB - 1) / EB, EB, 0, stream>>>(keys, idx, NN, j, k);
        gather_rows<<<(NN * 128 + EB - 1) / EB, EB, 0, stream>>>(t0, h, idx, NN);
        layernorm128<<<NN / 8, 256, 0, stream>>>(t0, t0, FP(base + 11), FP(base + 12), NN);
        run_mamba(stream, d_in, base, t0, t1, xz, xc, xdbl, dtb, yb);
        scatter_add_rows<<<(NN * 128 + EB - 1) / EB, EB, 0, stream>>>(h, t1, idx, NN);
    }

    // ---- higher-order fusion ----
    gemmF(stream, ADJHO, NN, h, 128, 0, nullptr, t0, 128, NN, NN, 128, ACT_NONE);
    gemmF(stream, h, 128, FP(HO + 0), 128, 1, FP(HO + 1), hn, 128, NN, 128, 128, ACT_NONE);
    gemmF(stream, t0, 128, FP(HO + 2), 128, 1, FP(HO + 3), hh, 128, NN, 128, 128, ACT_NONE);
    concat2<<<(NN * 256 + EB - 1) / EB, EB, 0, stream>>>(cat, hn, hh, NN);
    gemmF(stream, cat, 256, FP(HO + 4), 256, 1, FP(HO + 5), t1, 128, NN, 256, 128, ACT_NONE);
    gate_fuse<<<(NN * 128 + EB - 1) / EB, EB, 0, stream>>>(t0, t1, hn, hh, NN * 128);
    layernorm128<<<NN / 8, 256, 0, stream>>>(t0, h, FP(HO + 6), FP(HO + 7), NN);

    // ---- head ----
    gemmF(stream, h, 128, FP(75), 128, 1, FP(76), t0, 128, NN, 128, 128, ACT_RELU);
    gemmF(stream, t0, 128, FP(77), 128, 1, FP(78), t1, 64, NN, 128, 64, ACT_RELU);
    linear_naive<<<(NN + EB - 1) / EB, EB, 0, stream>>>(
        t1, 64, FP(79), FP(80), (float*)d_out, 1, NN, 64, 1, ACT_NONE);
}